// Encoder_72069551227429
// MI455X (gfx1250) — compile-verified
//
#include <hip/hip_runtime.h>

typedef __attribute__((ext_vector_type(2))) float v2f;
typedef __attribute__((ext_vector_type(8))) float v8f;

#define NUM_GRAPHS 500
#define DIM 128
#define BN_EPS 1e-5f

// ---------------------------------------------------------------------------
// zero fill
// ---------------------------------------------------------------------------
__global__ void zero_kernel(float* __restrict__ p, int n) {
    int i = blockIdx.x * 256 + threadIdx.x;
    if (i < n) p[i] = 0.0f;
}

// ---------------------------------------------------------------------------
// edge scatter: one wave32 per edge; lane loads float4 of the source row and
// atomically adds into aggr[dst]. Optionally accumulates degree (lane 0).
// Feature tables are L2-resident (51 MB each vs 192 MB L2), so these atomics
// and gathers run at L2 rates, not HBM.
// ---------------------------------------------------------------------------
__global__ __launch_bounds__(256) void scatter_kernel(
    const int* __restrict__ src, const int* __restrict__ dst,
    const float* __restrict__ feat, float* __restrict__ aggr,
    float* __restrict__ deg, int E)
{
    int e = blockIdx.x * 8 + (threadIdx.x >> 5);
    if (e >= E) return;
    int lane = threadIdx.x & 31;
    int s = src[e];
    int d = dst[e];
    float4 v = ((const float4*)(feat + (size_t)s * DIM))[lane];
    float* ar = aggr + (size_t)d * DIM + lane * 4;
    atomicAdd(ar + 0, v.x);
    atomicAdd(ar + 1, v.y);
    atomicAdd(ar + 2, v.z);
    atomicAdd(ar + 3, v.w);
    if (deg != nullptr && lane == 0) atomicAdd(deg + d, 1.0f);
}

// ---------------------------------------------------------------------------
// WMMA f32 GEMM:
//   out = relu( ((A1 [+ Aadd]) [* 1/max(rowscale,1)]) @ W1  [+ A2 @ W2] + bias )
// and, if stats != nullptr, accumulates per-feature sum / sum-of-squares of
// the output into stats[0..127] / stats[128..255] (for fused BatchNorm).
//
// Block: 256 threads = 8 waves; block covers 16 rows x 128 cols.
// Wave w computes the 16x16 tile at column 16*w with V_WMMA_F32_16X16X4_F32,
// K fully unrolled (32 wmma per pass). A tile staged in LDS (stride 132 to
// avoid 64-bank conflicts). Requires rows % 16 == 0 (N = 100000 = 6250*16).
// No divergence anywhere: EXEC is all-ones for WMMA and the shfl reduction.
// ---------------------------------------------------------------------------
__global__ __launch_bounds__(256) void gemm128_wmma(
    const float* __restrict__ A1, const float* __restrict__ Aadd,
    const float* __restrict__ rowscale,
    const float* __restrict__ W1,
    const float* __restrict__ A2, const float* __restrict__ W2,
    const float* __restrict__ bias, float* __restrict__ out,
    float* __restrict__ stats)
{
    __shared__ float As[16 * 132];
    const int tid  = threadIdx.x;
    const int wave = tid >> 5;
    const int lane = tid & 31;
    const int half = lane >> 4;   // 0: lanes 0-15, 1: lanes 16-31
    const int lr   = lane & 15;
    const int rowBase = blockIdx.x * 16;
    const int col     = wave * 16;

    v8f c = {};

    for (int pass = 0; pass < 2; ++pass) {
        const float* A = (pass == 0) ? A1 : A2;
        const float* W = (pass == 0) ? W1 : W2;
        if (A == nullptr) break;

        __syncthreads();
        for (int idx = tid; idx < 16 * DIM; idx += 256) {
            int r = idx >> 7, cc = idx & 127;
            float v = A[(size_t)(rowBase + r) * DIM + cc];
            if (pass == 0) {
                if (Aadd != nullptr)
                    v += Aadd[(size_t)(rowBase + r) * DIM + cc];
                if (rowscale != nullptr)
                    v /= fmaxf(rowscale[rowBase + r], 1.0f);
            }
            As[r * 132 + cc] = v;
        }
        __syncthreads();

        // ISA 32-bit fragment layouts:
        //  A 16x4 : lanes 0-15 -> M=lr, {K=k,k+1}; lanes 16-31 -> {K=k+2,k+3}
        //  B 4x16 : lanes 0-15 -> N=lr, {K=k,k+1}; lanes 16-31 -> {K=k+2,k+3}
        #pragma unroll
        for (int k = 0; k < DIM; k += 4) {
            int ka = k + half * 2;
            v2f a, b;
            a.x = As[lr * 132 + ka];
            a.y = As[lr * 132 + ka + 1];
            b.x = W[(size_t)ka * DIM + col + lr];
            b.y = W[(size_t)(ka + 1) * DIM + col + lr];
            c = __builtin_amdgcn_wmma_f32_16x16x4_f32(
                    false, a, false, b, (short)0, c, false, false);
        }
    }

    // C layout: VGPR j -> M=j (lanes 0-15) / M=j+8 (lanes 16-31), N = lr
    float bv = bias[col + lr];
    float s = 0.0f, ss = 0.0f;
    #pragma unroll
    for (int j = 0; j < 8; ++j) {
        float v = c[j] + bv;
        v = v > 0.0f ? v : 0.0f;
        out[(size_t)(rowBase + half * 8 + j) * DIM + col + lr] = v;
        s  += v;
        ss += v * v;
    }

    if (stats != nullptr) {
        // combine the two half-wave partners holding the same column
        s  += __shfl_xor(s, 16);
        ss += __shfl_xor(ss, 16);
        if (half == 0) {
            atomicAdd(&stats[col + lr], s);
            atomicAdd(&stats[DIM + col + lr], ss);
        }
    }
}

// ---------------------------------------------------------------------------
// BatchNorm finalize: stats[0..127]=sum, [128..255]=sumsq  ->
//                     stats[256..383]=scale, [384..511]=shift
// ---------------------------------------------------------------------------
__global__ void bn_finalize_kernel(float* __restrict__ stats,
                                   const float* __restrict__ gamma,
                                   const float* __restrict__ beta, int N)
{
    int f = threadIdx.x;
    if (f >= DIM) return;
    float inv_n = 1.0f / (float)N;
    float mu  = stats[f] * inv_n;
    float var = stats[DIM + f] * inv_n - mu * mu;
    float sc  = gamma[f] * rsqrtf(var + BN_EPS);
    stats[2 * DIM + f] = sc;                // scale
    stats[3 * DIM + f] = beta[f] - mu * sc; // shift
}

// ---------------------------------------------------------------------------
// BN apply + write node_feats slice + global_add_pool atomics + keep h for
// next layer.
// ---------------------------------------------------------------------------
__global__ __launch_bounds__(256) void bn_apply_kernel(
    const float* __restrict__ pre, const float* __restrict__ stats,
    const int* __restrict__ batch,
    float* __restrict__ out_nodes, float* __restrict__ out_pool,
    float* __restrict__ hout, int layer, int total)
{
    int idx = blockIdx.x * 256 + threadIdx.x;
    if (idx >= total) return;
    int i = idx >> 7, f = idx & 127;
    float v = pre[idx] * stats[2 * DIM + f] + stats[3 * DIM + f];
    hout[idx] = v;
    out_nodes[(size_t)i * (3 * DIM) + layer * DIM + f] = v;
    atomicAdd(&out_pool[(size_t)batch[i] * (3 * DIM) + layer * DIM + f], v);
}

// ---------------------------------------------------------------------------
extern "C" void kernel_launch(void* const* d_in, const int* in_sizes, int n_in,
                              void* d_out, int out_size, void* d_ws, size_t ws_size,
                              hipStream_t stream)
{
    const float* x        = (const float*)d_in[0];
    const int*   ei       = (const int*)d_in[1];
    const int*   batch    = (const int*)d_in[2];
    const float* sage_Wl  = (const float*)d_in[3];
    const float* sage_bl  = (const float*)d_in[4];
    const float* sage_Wr  = (const float*)d_in[5];
    const float* gin_W1   = (const float*)d_in[6];
    const float* gin_b1   = (const float*)d_in[7];
    const float* gin_W2   = (const float*)d_in[8];
    const float* gin_b2   = (const float*)d_in[9];
    const float* bn_gamma = (const float*)d_in[10];
    const float* bn_beta  = (const float*)d_in[11];

    const int N = in_sizes[0] / DIM;     // 100000
    const int E = in_sizes[1] / 2;       // 1600000
    const int ND = N * DIM;              // 12.8M

    const int* src = ei;
    const int* dst = ei + E;

    // workspace layout
    float* buf0  = (float*)d_ws;           // aggr / gin gemm2 output
    float* buf1  = buf0 + (size_t)ND;      // h (post-BN activations)
    float* buf2  = buf1 + (size_t)ND;      // gemm scratch
    float* deg   = buf2 + (size_t)ND;      // N
    float* stats = deg + N;                // 512 floats

    float* out_pool  = (float*)d_out;                    // (500, 384)
    float* out_nodes = out_pool + (size_t)NUM_GRAPHS * 3 * DIM;

    const int gridND   = (ND + 255) / 256;
    const int gridE    = (E + 7) / 8;
    const int rowTiles = N / 16;           // N divisible by 16

    // zero accumulators (re-zeroed every call: graph-replay safe)
    zero_kernel<<<(NUM_GRAPHS * 3 * DIM + 255) / 256, 256, 0, stream>>>(
        out_pool, NUM_GRAPHS * 3 * DIM);
    zero_kernel<<<(N + 255) / 256, 256, 0, stream>>>(deg, N);
    zero_kernel<<<gridND, 256, 0, stream>>>(buf0, ND);

    // ---- Layer 0: SAGEConv ----
    scatter_kernel<<<gridE, 256, 0, stream>>>(src, dst, x, buf0, deg, E);
    zero_kernel<<<2, 256, 0, stream>>>(stats, 512);
    // h_pre = relu((aggr/deg)@Wl + x@Wr + bl) -> buf2 ; BN stats fused
    gemm128_wmma<<<rowTiles, 256, 0, stream>>>(
        buf0, nullptr, deg, sage_Wl, x, sage_Wr, sage_bl, buf2, stats);
    bn_finalize_kernel<<<1, 128, 0, stream>>>(stats, bn_gamma, bn_beta, N);
    bn_apply_kernel<<<gridND, 256, 0, stream>>>(
        buf2, stats, batch, out_nodes, out_pool, buf1, 0, ND);

    // ---- Layers 1..2: GINConv ----
    for (int l = 1; l < 3; ++l) {
        const float* W1 = gin_W1 + (size_t)(l - 1) * DIM * DIM;
        const float* b1 = gin_b1 + (size_t)(l - 1) * DIM;
        const float* W2 = gin_W2 + (size_t)(l - 1) * DIM * DIM;
        const float* b2 = gin_b2 + (size_t)(l - 1) * DIM;

        zero_kernel<<<gridND, 256, 0, stream>>>(buf0, ND);
        scatter_kernel<<<gridE, 256, 0, stream>>>(src, dst, buf1, buf0, nullptr, E);
        // t1 = relu((h + aggr) @ W1 + b1) -> buf2   (no BN stats)
        gemm128_wmma<<<rowTiles, 256, 0, stream>>>(
            buf1, buf0, nullptr, W1, nullptr, nullptr, b1, buf2, nullptr);
        zero_kernel<<<2, 256, 0, stream>>>(stats, 512);
        // h_pre = relu(t1 @ W2 + b2) -> buf0 ; BN stats fused
        gemm128_wmma<<<rowTiles, 256, 0, stream>>>(
            buf2, nullptr, nullptr, W2, nullptr, nullptr, b2, buf0, stats);
        bn_finalize_kernel<<<1, 128, 0, stream>>>(
            stats, bn_gamma + (size_t)l * DIM, bn_beta + (size_t)l * DIM, N);
        bn_apply_kernel<<<gridND, 256, 0, stream>>>(
            buf0, stats, batch, out_nodes, out_pool, buf1, l, ND);
    }
}